// TropicalConv2D_67242007986179
// MI455X (gfx1250) — compile-verified
//
#include <hip/hip_runtime.h>
#include <math.h>

typedef unsigned int u32;
typedef unsigned long long u64;
typedef u32 v4u __attribute__((ext_vector_type(4)));
typedef u32 v8u __attribute__((ext_vector_type(8)));

// Problem constants (from reference setup_inputs)
#define B_  8
#define C_  16
#define H_  64
#define W_  64
#define O_  32
#define R_  144          // C_*3*3
#define TILE_H 8
#define ROWS   10        // TILE_H + 2 halo rows
#define ROWW   66        // 64 cols + 2 halo cols (TDM pad_amount gives the +2)
#define CH_STRIDE (ROWS * ROWW)       // 660 floats per channel slab
#define LDSX (C_ * CH_STRIDE)         // 10560 floats
#define LDSW (R_ * O_)                // 4608 floats, transposed: [r][oc]

__global__ __launch_bounds__(256)
void TropicalConv2D_kernel(const float* __restrict__ x,
                           const float* __restrict__ w,
                           float* __restrict__ out) {
  __shared__ __align__(16) float sx[LDSX];
  __shared__ __align__(16) float swt[LDSW];   // swt[r*32 + oc] = w[oc*144 + r]

  const int tid = threadIdx.x;
  const int b   = blockIdx.x >> 3;
  const int h0  = (blockIdx.x & 7) * TILE_H;

  // ---- Phase 1: zero x-tile (halo cells must read 0 == zero padding),
  //               stage weights into LDS transposed to [r][oc] so the hot
  //               loop can fetch 32 weights with 8x ds_load_b128 broadcasts.
  for (int i = tid; i < LDSX; i += 256) sx[i] = 0.0f;
  for (int i = tid; i < LDSW; i += 256) {
    const int oc = i / R_;        // coalesced global read ...
    const int r  = i % R_;
    swt[r * O_ + oc] = w[i];      // ... scattered LDS write (one-time, 18 KB)
  }
  __syncthreads();   // ds stores complete before TDM writes the same LDS region

  // ---- Phase 2: Tensor Data Mover: each wave DMAs 2 input channels
  //               (valid rows only; halo rows/cols remain zero).
  {
    const int wave   = tid >> 5;                       // 0..7, wave-uniform
    const int rStart = (h0 == 0) ? 0 : (h0 - 1);
    const int rEnd   = (h0 + TILE_H >= H_) ? (H_ - 1) : (h0 + TILE_H);
    const int nvalid = rEnd - rStart + 1;              // 9 at edges, 10 interior
    const int rowOff = rStart - (h0 - 1);              // 1 at top edge, else 0
    const u32 ldsBase = (u32)(size_t)(&sx[0]);         // low 32 bits of generic ptr = LDS offset

    // D# group 1 (CDNA5 ISA), channel-independent parts:
    //   workgroup_mask=0 (not in cluster), data_size=2 (4B),
    //   pad_enable=1, pad_interval=5 (every 64 DWORDs), pad_amount=1 (2 DWORDs)
    //   -> LDS rows become 66 DWORDs wide, matching the halo layout.
    v8u g1;
    g1[0] = (2u << 16) | (1u << 20) | (5u << 22) | (1u << 25);
    g1[1] = (u32)W_ << 16;                      // tensor_dim0 = 64 (bits 63:48)
    g1[2] = ((u32)nvalid & 0xFFFFu) << 16;      // tensor_dim1[15:0] (bits 95:80)
    g1[3] = (u32)W_ << 16;                      // tile_dim0 = 64 (bits 127:112)
    g1[4] = (u32)nvalid;                        // tile_dim1 = valid rows (bits 143:128)
    g1[5] = (u32)W_;                            // tensor_dim0_stride = 64 (bits 207:160)
    g1[6] = 0u;                                 // tensor_dim1_stride unused (2-D tile)
    g1[7] = 0u;

    for (int cc = 0; cc < 2; ++cc) {
      const int c = wave * 2 + cc;
      const u64 gaddr = (u64)(size_t)(x + ((u64)(b * C_ + c) * (H_ * W_)
                                           + (u64)rStart * W_));
      const u32 laddr = ldsBase + (u32)((c * CH_STRIDE + rowOff * ROWW + 1) * 4);

      // D# group 0 (ISA §8.3): count=1, lds_addr, global_addr[56:0], type=2.
      v4u g0;
      g0[0] = 1u;
      g0[1] = laddr;
      g0[2] = (u32)gaddr;
      g0[3] = ((u32)(gaddr >> 32) & 0x01FFFFFFu) | (2u << 30);

      // 2-group form (VADDR2/3 = NULL): tensors up to 2D.
      asm volatile("tensor_load_to_lds %0, %1"
                   :: "s"(g0), "s"(g1)
                   : "memory");
    }
    __builtin_amdgcn_s_wait_tensorcnt(0);   // each wave waits on its own TDM ops
  }
  __syncthreads();

  // ---- Phase 3: compute. Each thread: 2 adjacent pixels x 32 output channels.
  const int row = tid >> 5;            // 0..7  (pix0 = 2*tid)
  const int col = (tid & 31) * 2;      // 0,2,...,62

  float acc0[O_], acc1[O_];
#pragma unroll
  for (int oc = 0; oc < O_; ++oc) { acc0[oc] = -INFINITY; acc1[oc] = -INFINITY; }

  for (int c = 0; c < C_; ++c) {
    const float* __restrict__ xc = &sx[c * CH_STRIDE + row * ROWW + col];
    const float* __restrict__ wr = &swt[c * 9 * O_];   // taps of this channel
#pragma unroll
    for (int di = 0; di < 3; ++di) {
      // 4 contiguous pixels cover dj=0..2 for both output pixels (col, col+1)
      float p[4];
#pragma unroll
      for (int k = 0; k < 4; ++k) p[k] = xc[di * ROWW + k];
#pragma unroll
      for (int dj = 0; dj < 3; ++dj) {
        const float p0 = p[dj];
        const float p1 = p[dj + 1];
        const float4* __restrict__ wv4 =
            reinterpret_cast<const float4*>(&wr[(di * 3 + dj) * O_]);
#pragma unroll
        for (int q = 0; q < 8; ++q) {            // 8x ds_load_b128 (broadcast)
          const float4 wv = wv4[q];
          acc0[q * 4 + 0] = fmaxf(acc0[q * 4 + 0], p0 - wv.x);
          acc0[q * 4 + 1] = fmaxf(acc0[q * 4 + 1], p0 - wv.y);
          acc0[q * 4 + 2] = fmaxf(acc0[q * 4 + 2], p0 - wv.z);
          acc0[q * 4 + 3] = fmaxf(acc0[q * 4 + 3], p0 - wv.w);
          acc1[q * 4 + 0] = fmaxf(acc1[q * 4 + 0], p1 - wv.x);
          acc1[q * 4 + 1] = fmaxf(acc1[q * 4 + 1], p1 - wv.y);
          acc1[q * 4 + 2] = fmaxf(acc1[q * 4 + 2], p1 - wv.z);
          acc1[q * 4 + 3] = fmaxf(acc1[q * 4 + 3], p1 - wv.w);
        }
      }
    }
  }

  const int hr = h0 + row;
#pragma unroll
  for (int oc = 0; oc < O_; ++oc) {
    float2 v = make_float2(acc0[oc], acc1[oc]);
    *reinterpret_cast<float2*>(
        &out[(((u64)(b * O_ + oc) * H_) + (u64)hr) * W_ + col]) = v;  // b64 store
  }
}

extern "C" void kernel_launch(void* const* d_in, const int* in_sizes, int n_in,
                              void* d_out, int out_size, void* d_ws, size_t ws_size,
                              hipStream_t stream) {
  (void)in_sizes; (void)n_in; (void)d_ws; (void)ws_size; (void)out_size;
  const float* x = (const float*)d_in[0];
  const float* w = (const float*)d_in[1];
  float* out = (float*)d_out;
  dim3 grid(B_ * (H_ / TILE_H));   // 64 workgroups
  dim3 block(256);                 // 8 wave32s
  hipLaunchKernelGGL(TropicalConv2D_kernel, grid, block, 0, stream, x, w, out);
}